// GAT_738734375587
// MI455X (gfx1250) — compile-verified
//
#include <hip/hip_runtime.h>
#include <math.h>

typedef __attribute__((ext_vector_type(2))) float v2f;
typedef __attribute__((ext_vector_type(8))) float v8f;

#define F_IN 128
#define HC1  64      // H * C1
#define NHEAD 8
#define CH1   8
#define NC    40
#define NEG_SLOPE 0.2f

// ---- monotone float <-> uint encoding for atomicMax on floats ----
__device__ __forceinline__ unsigned f2ord(float f) {
    unsigned u = __float_as_uint(f);
    return (u & 0x80000000u) ? ~u : (u | 0x80000000u);
}
__device__ __forceinline__ float ord2f(unsigned k) {
    return (k & 0x80000000u) ? __uint_as_float(k & 0x7FFFFFFFu)
                             : __uint_as_float(~k);
}
__device__ __forceinline__ float leaky(float x) { return x > 0.0f ? x : NEG_SLOPE * x; }

// ---------------- zero fill ----------------
__global__ void gat_zero(float* __restrict__ p, int count) {
    int i = blockIdx.x * blockDim.x + threadIdx.x;
    for (; i < count; i += gridDim.x * blockDim.x) p[i] = 0.0f;
}

// ---------------- GEMM1: feats1[N,64] = x[N,128] @ W1[128,64] (WMMA f32 16x16x4) ----
// W is staged in LDS as K-pairs: sW2[kp*64+col] = {W[2kp][col], W[2kp+1][col]}
// so each B fragment is ONE aligned ds_load_b64 straight into the operand pair.
__global__ void gat_gemm1(const float* __restrict__ x, const float* __restrict__ W,
                          float* __restrict__ feats, int n) {
    __shared__ v2f sW2[(F_IN / 2) * HC1];            // 32 KB
    for (int i = threadIdx.x; i < (F_IN / 2) * HC1; i += blockDim.x) {
        const int kp = i >> 6, col = i & 63;
        v2f w; w.x = W[(2 * kp) * HC1 + col]; w.y = W[(2 * kp + 1) * HC1 + col];
        sW2[i] = w;
    }
    __syncthreads();
    const int wave = threadIdx.x >> 5, lane = threadIdx.x & 31;
    const int tile = blockIdx.x * (blockDim.x >> 5) + wave;
    const int ntiles = (n + 15) >> 4;
    if (tile >= ntiles) return;
    const int half = lane >> 4;      // 0: lanes 0-15, 1: lanes 16-31
    const int l16  = lane & 15;
    int row = tile * 16 + l16; if (row >= n) row = n - 1;   // clamp (tail-safe)
    const float* xrow = x + (size_t)row * F_IN;

    v8f acc0 = {}, acc1 = {}, acc2 = {}, acc3 = {};
    #pragma unroll 4
    for (int kk = 0; kk < F_IN / 4; ++kk) {
        const int ka = kk * 4 + half * 2;           // A: VGPR0=K(ka), VGPR1=K(ka+1)
        v2f a; a.x = xrow[ka]; a.y = xrow[ka + 1];
        const v2f* wrow = sW2 + (size_t)(kk * 2 + half) * HC1 + l16;
        v2f b0 = wrow[0], b1 = wrow[16], b2 = wrow[32], b3 = wrow[48];
        acc0 = __builtin_amdgcn_wmma_f32_16x16x4_f32(false, a, false, b0, (short)0, acc0, false, false);
        acc1 = __builtin_amdgcn_wmma_f32_16x16x4_f32(false, a, false, b1, (short)0, acc1, false, false);
        acc2 = __builtin_amdgcn_wmma_f32_16x16x4_f32(false, a, false, b2, (short)0, acc2, false, false);
        acc3 = __builtin_amdgcn_wmma_f32_16x16x4_f32(false, a, false, b3, (short)0, acc3, false, false);
    }
    #pragma unroll
    for (int r = 0; r < 8; ++r) {                   // D: VGPR r -> M=r (+8 for hi half)
        const int orow = tile * 16 + r + half * 8;
        if (orow < n) {
            float* o = feats + (size_t)orow * HC1 + l16;
            o[ 0] = acc0[r]; o[16] = acc1[r]; o[32] = acc2[r]; o[48] = acc3[r];
        }
    }
}

// ---------------- per-(node,head) attention coefficients, layer 1 ----------------
__global__ void gat_alpha1(const float* __restrict__ feats, const float* __restrict__ as,
                           const float* __restrict__ ad, float* __restrict__ alpha_s,
                           float* __restrict__ alpha_d, int n) {
    int idx = blockIdx.x * blockDim.x + threadIdx.x;
    if (idx >= n * NHEAD) return;
    const int node = idx >> 3, h = idx & 7;
    const float* f = feats + (size_t)node * HC1 + h * CH1;
    const float* s = as + h * CH1;
    const float* d = ad + h * CH1;
    float xs = 0.f, xd = 0.f;
    #pragma unroll
    for (int c = 0; c < CH1; ++c) { float v = f[c]; xs = fmaf(v, s[c], xs); xd = fmaf(v, d[c], xd); }
    alpha_s[idx] = xs; alpha_d[idx] = xd;
}

// ---------------- layer-1 edge passes (per edge x head) ----------------
__device__ __forceinline__ void edge_sd(const int* ei, int E, int e, int& s, int& d) {
    if (e < E) { s = ei[e]; d = ei[E + e]; } else { s = d = e - E; }  // self-loops appended
}

__global__ void gat_emax1(const int* __restrict__ ei, int E, int n,
                          const float* __restrict__ as, const float* __restrict__ ad,
                          unsigned* __restrict__ mkey) {
    int idx = blockIdx.x * blockDim.x + threadIdx.x;
    if (idx >= (E + n) * NHEAD) return;
    int e = idx >> 3, h = idx & 7, s, d; edge_sd(ei, E, e, s, d);
    float ev = leaky(as[s * NHEAD + h] + ad[d * NHEAD + h]);
    atomicMax(&mkey[d * NHEAD + h], f2ord(ev));
}

__global__ void gat_eden1(const int* __restrict__ ei, int E, int n,
                          const float* __restrict__ as, const float* __restrict__ ad,
                          const unsigned* __restrict__ mkey, float* __restrict__ den) {
    int idx = blockIdx.x * blockDim.x + threadIdx.x;
    if (idx >= (E + n) * NHEAD) return;
    int e = idx >> 3, h = idx & 7, s, d; edge_sd(ei, E, e, s, d);
    float ev = leaky(as[s * NHEAD + h] + ad[d * NHEAD + h]);
    atomicAdd(&den[d * NHEAD + h], __expf(ev - ord2f(mkey[d * NHEAD + h])));
}

__global__ void gat_eagg1(const int* __restrict__ ei, int E, int n,
                          const float* __restrict__ as, const float* __restrict__ ad,
                          const unsigned* __restrict__ mkey, const float* __restrict__ den,
                          const float* __restrict__ feats, float* __restrict__ out) {
    int idx = blockIdx.x * blockDim.x + threadIdx.x;
    if (idx >= (E + n) * NHEAD) return;
    int e = idx >> 3, h = idx & 7, s, d; edge_sd(ei, E, e, s, d);
    float ev = leaky(as[s * NHEAD + h] + ad[d * NHEAD + h]);
    float w = __expf(ev - ord2f(mkey[d * NHEAD + h])) / fmaxf(den[d * NHEAD + h], 1e-16f);
    const float* fs = feats + (size_t)s * HC1 + h * CH1;
    float* od = out + (size_t)d * HC1 + h * CH1;
    #pragma unroll
    for (int c = 0; c < CH1; ++c) atomicAdd(&od[c], w * fs[c]);
}

// ---------------- bias + relu, in place ----------------
__global__ void gat_bias_relu(float* __restrict__ h, const float* __restrict__ b, int n) {
    int idx = blockIdx.x * blockDim.x + threadIdx.x;
    if (idx >= n * HC1) return;
    h[idx] = fmaxf(h[idx] + b[idx & (HC1 - 1)], 0.0f);
}

// ---------------- GEMM2: feats2[N,40] = h1[N,64] @ W2[64,40] (WMMA f32 16x16x4) ----
__global__ void gat_gemm2(const float* __restrict__ h1, const float* __restrict__ W,
                          float* __restrict__ feats, int n) {
    __shared__ v2f sW2[(HC1 / 2) * NC];              // 10 KB, K-paired
    for (int i = threadIdx.x; i < (HC1 / 2) * NC; i += blockDim.x) {
        const int kp = i / NC, col = i - kp * NC;
        v2f w; w.x = W[(2 * kp) * NC + col]; w.y = W[(2 * kp + 1) * NC + col];
        sW2[i] = w;
    }
    __syncthreads();
    const int wave = threadIdx.x >> 5, lane = threadIdx.x & 31;
    const int tile = blockIdx.x * (blockDim.x >> 5) + wave;
    const int ntiles = (n + 15) >> 4;
    if (tile >= ntiles) return;
    const int half = lane >> 4, l16 = lane & 15;
    int row = tile * 16 + l16; if (row >= n) row = n - 1;
    const float* hrow = h1 + (size_t)row * HC1;

    v8f acc0 = {}, acc1 = {}, acc2 = {};
    const bool c2ok = (l16 + 32) < NC;
    const v2f bz = {0.f, 0.f};
    #pragma unroll 4
    for (int kk = 0; kk < HC1 / 4; ++kk) {
        const int ka = kk * 4 + half * 2;
        v2f a; a.x = hrow[ka]; a.y = hrow[ka + 1];
        const v2f* wrow = sW2 + (size_t)(kk * 2 + half) * NC;
        v2f b0 = wrow[l16], b1 = wrow[l16 + 16];
        v2f b2 = c2ok ? wrow[l16 + 32] : bz;
        acc0 = __builtin_amdgcn_wmma_f32_16x16x4_f32(false, a, false, b0, (short)0, acc0, false, false);
        acc1 = __builtin_amdgcn_wmma_f32_16x16x4_f32(false, a, false, b1, (short)0, acc1, false, false);
        acc2 = __builtin_amdgcn_wmma_f32_16x16x4_f32(false, a, false, b2, (short)0, acc2, false, false);
    }
    #pragma unroll
    for (int r = 0; r < 8; ++r) {
        const int orow = tile * 16 + r + half * 8;
        if (orow < n) {
            float* o = feats + (size_t)orow * NC;
            o[l16]      = acc0[r];
            o[l16 + 16] = acc1[r];
            if (c2ok) o[l16 + 32] = acc2[r];
        }
    }
}

// ---------------- layer-2 alpha (H=1, 40 channels) ----------------
__global__ void gat_alpha2(const float* __restrict__ feats, const float* __restrict__ as,
                           const float* __restrict__ ad, float* __restrict__ alpha_s,
                           float* __restrict__ alpha_d, int n) {
    int node = blockIdx.x * blockDim.x + threadIdx.x;
    if (node >= n) return;
    const float* f = feats + (size_t)node * NC;
    float xs = 0.f, xd = 0.f;
    #pragma unroll
    for (int c = 0; c < NC; ++c) { float v = f[c]; xs = fmaf(v, as[c], xs); xd = fmaf(v, ad[c], xd); }
    alpha_s[node] = xs; alpha_d[node] = xd;
}

// ---------------- layer-2 edge passes ----------------
__global__ void gat_emax2(const int* __restrict__ ei, int E, int n,
                          const float* __restrict__ as, const float* __restrict__ ad,
                          unsigned* __restrict__ mkey) {
    int e = blockIdx.x * blockDim.x + threadIdx.x;
    if (e >= E + n) return;
    int s, d; edge_sd(ei, E, e, s, d);
    atomicMax(&mkey[d], f2ord(leaky(as[s] + ad[d])));
}

__global__ void gat_eden2(const int* __restrict__ ei, int E, int n,
                          const float* __restrict__ as, const float* __restrict__ ad,
                          const unsigned* __restrict__ mkey, float* __restrict__ den) {
    int e = blockIdx.x * blockDim.x + threadIdx.x;
    if (e >= E + n) return;
    int s, d; edge_sd(ei, E, e, s, d);
    float ev = leaky(as[s] + ad[d]);
    atomicAdd(&den[d], __expf(ev - ord2f(mkey[d])));
}

__global__ void gat_eagg2(const int* __restrict__ ei, int E, int n,
                          const float* __restrict__ as, const float* __restrict__ ad,
                          const unsigned* __restrict__ mkey, const float* __restrict__ den,
                          const float* __restrict__ feats, float* __restrict__ out) {
    int idx = blockIdx.x * blockDim.x + threadIdx.x;          // (edge, group-of-8) pairs
    if (idx >= (E + n) * 5) return;
    int e = idx / 5, g = idx - e * 5;
    int s, d; edge_sd(ei, E, e, s, d);
    float ev = leaky(as[s] + ad[d]);
    float w = __expf(ev - ord2f(mkey[d])) / fmaxf(den[d], 1e-16f);
    const float* fs = feats + (size_t)s * NC + g * 8;
    float* od = out + (size_t)d * NC + g * 8;
    #pragma unroll
    for (int c = 0; c < 8; ++c) atomicAdd(&od[c], w * fs[c]);
}

// ---------------- final bias + log_softmax ----------------
__global__ void gat_logsoftmax(const float* __restrict__ h, const float* __restrict__ b,
                               float* __restrict__ out, int n) {
    int node = blockIdx.x * blockDim.x + threadIdx.x;
    if (node >= n) return;
    const float* r = h + (size_t)node * NC;
    float v[NC], mx = -INFINITY;
    #pragma unroll
    for (int c = 0; c < NC; ++c) { v[c] = r[c] + b[c]; mx = fmaxf(mx, v[c]); }
    float sum = 0.f;
    #pragma unroll
    for (int c = 0; c < NC; ++c) sum += __expf(v[c] - mx);
    float ls = __logf(sum);
    float* o = out + (size_t)node * NC;
    #pragma unroll
    for (int c = 0; c < NC; ++c) o[c] = v[c] - mx - ls;
}

// =======================================================================
extern "C" void kernel_launch(void* const* d_in, const int* in_sizes, int n_in,
                              void* d_out, int out_size, void* d_ws, size_t ws_size,
                              hipStream_t stream) {
    const float* x   = (const float*)d_in[0];
    const int*   ei  = (const int*)  d_in[1];
    const float* W1  = (const float*)d_in[2];
    const float* as1 = (const float*)d_in[3];
    const float* ad1 = (const float*)d_in[4];
    const float* b1  = (const float*)d_in[5];
    const float* W2  = (const float*)d_in[6];
    const float* as2 = (const float*)d_in[7];
    const float* ad2 = (const float*)d_in[8];
    const float* b2  = (const float*)d_in[9];

    const int n = in_sizes[0] / F_IN;       // 50000
    const int E = in_sizes[1] / 2;          // 1.6M
    const int Etot = E + n;

    // workspace layout (floats); regions reused between layers (sequential stream order)
    float*    ws      = (float*)d_ws;
    float*    feats1  = ws;                          // n*64; reused as feats2 (n*40)
    float*    alpha_s = ws + (size_t)n * 64;         // n*8 (L1) / n (L2)
    float*    alpha_d = ws + (size_t)n * 72;         // n*8 (L1) / n (L2)
    unsigned* mkey    = (unsigned*)(ws + (size_t)n * 80); // n*8 (L1) / n (L2)
    float*    den     = ws + (size_t)n * 88;         // n*8 (L1) / n (L2)
    float*    out1    = ws + (size_t)n * 96;         // n*64 accumulator -> h1 (in-place)
    float*    out2    = ws + (size_t)n * 160;        // n*40 accumulator

    const int B = 256;
    const int gemmBlocks = (((n + 15) / 16) + 7) / 8;          // 8 waves/block
    auto cdiv = [](long long a, int b) { return (int)((a + b - 1) / b); };

    // ---- layer 1 ----
    gat_zero<<<cdiv((long long)n * 80, B), B, 0, stream>>>(ws + (size_t)n * 80, n * 80); // m,den,out1
    gat_gemm1<<<gemmBlocks, B, 0, stream>>>(x, W1, feats1, n);
    gat_alpha1<<<cdiv((long long)n * NHEAD, B), B, 0, stream>>>(feats1, as1, ad1, alpha_s, alpha_d, n);
    gat_emax1<<<cdiv((long long)Etot * NHEAD, B), B, 0, stream>>>(ei, E, n, alpha_s, alpha_d, mkey);
    gat_eden1<<<cdiv((long long)Etot * NHEAD, B), B, 0, stream>>>(ei, E, n, alpha_s, alpha_d, mkey, den);
    gat_eagg1<<<cdiv((long long)Etot * NHEAD, B), B, 0, stream>>>(ei, E, n, alpha_s, alpha_d, mkey, den, feats1, out1);
    gat_bias_relu<<<cdiv((long long)n * HC1, B), B, 0, stream>>>(out1, b1, n);  // out1 -> h1

    // ---- layer 2 ----
    gat_zero<<<cdiv((long long)n * 16, B), B, 0, stream>>>(ws + (size_t)n * 80, n * 16); // m2,den2
    gat_zero<<<cdiv((long long)n * NC, B), B, 0, stream>>>(out2, n * NC);
    gat_gemm2<<<gemmBlocks, B, 0, stream>>>(out1, W2, feats1, n);                // feats2 := feats1 region
    gat_alpha2<<<cdiv(n, B), B, 0, stream>>>(feats1, as2, ad2, alpha_s, alpha_d, n);
    gat_emax2<<<cdiv(Etot, B), B, 0, stream>>>(ei, E, n, alpha_s, alpha_d, mkey);
    gat_eden2<<<cdiv(Etot, B), B, 0, stream>>>(ei, E, n, alpha_s, alpha_d, mkey, den);
    gat_eagg2<<<cdiv((long long)Etot * 5, B), B, 0, stream>>>(ei, E, n, alpha_s, alpha_d, mkey, den, feats1, out2);
    gat_logsoftmax<<<cdiv(n, B), B, 0, stream>>>(out2, b2, (float*)d_out, n);
}